// MyModel_71768903516316
// MI455X (gfx1250) — compile-verified
//
#include <hip/hip_runtime.h>
#include <math.h>

typedef __attribute__((ext_vector_type(2))) float v2f;
typedef __attribute__((ext_vector_type(8))) float v8f;

// ---------------------------------------------------------------------------
// Bilinear 1-D horizontal sample with zeros padding (matches jax grid_sample
// style warp in the reference: v0/v1 validity masks, clipped gathers).
// ---------------------------------------------------------------------------
__device__ __forceinline__ float sample1d(const float* __restrict__ row, int W, float px) {
  float x0f = floorf(px);
  float wx  = px - x0f;
  int x0 = (int)x0f;
  int x1 = x0 + 1;
  float v0 = (x0 >= 0 && x0 <= W - 1) ? 1.f : 0.f;
  float v1 = (x1 >= 0 && x1 <= W - 1) ? 1.f : 0.f;
  int x0c = x0 < 0 ? 0 : (x0 > W - 1 ? W - 1 : x0);
  int x1c = x1 < 0 ? 0 : (x1 > W - 1 ? W - 1 : x1);
  return row[x0c] * ((1.f - wx) * v0) + row[x1c] * (wx * v1);
}

__device__ __forceinline__ v8f wmma_f32_16x16x4(v2f a, v2f b, v8f c) {
  // (neg_a, A, neg_b, B, c_mod, C, reuse_a, reuse_b)
  return __builtin_amdgcn_wmma_f32_16x16x4_f32(false, a, false, b, (short)0, c,
                                               false, false);
}

// ---------------------------------------------------------------------------
// Zero the accumulator slots (40 used, 64 reserved).
// ---------------------------------------------------------------------------
__global__ void init_kernel(float* __restrict__ acc) {
  acc[threadIdx.x] = 0.f;
}

// ---------------------------------------------------------------------------
// Pyramid level: direct 2^l x 2^l block mean of the base image (identical to
// iterated 2x2 average pooling).
// ---------------------------------------------------------------------------
__global__ void __launch_bounds__(256) pyr_kernel(const float* __restrict__ src,
                                                  float* __restrict__ dst,
                                                  int s, int h, int w, int N) {
  int idx = blockIdx.x * 256 + threadIdx.x;
  if (idx >= N) return;
  int x = idx % w;
  int y = (idx / w) % h;
  int c = (idx / (w * h)) % 3;
  int b = idx / (w * h * 3);
  int H0 = h * s, W0 = w * s;
  const float* p = src + (((size_t)b * 3 + c) * H0 + (size_t)y * s) * W0 + (size_t)x * s;
  float sum = 0.f;
  for (int dy = 0; dy < s; ++dy)
    for (int dx = 0; dx < s; ++dx)
      sum += p[(size_t)dy * W0 + dx];
  dst[idx] = sum / (float)(s * s);
}

// ---------------------------------------------------------------------------
// Per-pixel fused kernel: L1 reconstruction, LR consistency, edge-weighted
// smoothness. Accumulates 8 sums via block reduction + float atomics.
// acc slots: 0 l1_l, 1 l1_r, 4 smx_dl, 5 smy_dl, 6 smx_dr, 7 smy_dr,
//            8 lr_l, 9 lr_r
// ---------------------------------------------------------------------------
__global__ void __launch_bounds__(256) warp_loss_kernel(
    const float* __restrict__ Lp, const float* __restrict__ Rp,
    const float* __restrict__ disp, float* __restrict__ acc,
    int H, int W, int N) {
  __shared__ float red[256];
  int tid = threadIdx.x;
  int idx = blockIdx.x * 256 + tid;
  float p0 = 0.f, p1 = 0.f, p2 = 0.f, p3 = 0.f, p4 = 0.f, p5 = 0.f, p6 = 0.f, p7 = 0.f;
  if (idx < N) {
    int x = idx % W;
    int y = (idx / W) % H;
    int b = idx / (W * H);
    size_t hw = (size_t)H * W;
    const float* dl = disp + (size_t)b * 2 * hw + (size_t)y * W;
    const float* dr = dl + hw;
    float dlv = dl[x], drv = dr[x];
    float pxl = (float)x - dlv * (float)(W - 1);
    float pxr = (float)x + drv * (float)(W - 1);
    const float* Lb = Lp + (size_t)b * 3 * hw + (size_t)y * W;
    const float* Rb = Rp + (size_t)b * 3 * hw + (size_t)y * W;
    float gxl = 0.f, gxr = 0.f, gyl = 0.f, gyr = 0.f;
    for (int c = 0; c < 3; ++c) {
      const float* Lr = Lb + (size_t)c * hw;
      const float* Rr = Rb + (size_t)c * hw;
      float le = sample1d(Rr, W, pxl);   // left_est = gen_left(R, dl)
      float re = sample1d(Lr, W, pxr);   // right_est = gen_right(L, dr)
      p0 += fabsf(le - Lr[x]);
      p1 += fabsf(re - Rr[x]);
      if (x < W - 1) { gxl += fabsf(Lr[x] - Lr[x + 1]); gxr += fabsf(Rr[x] - Rr[x + 1]); }
      if (y < H - 1) { gyl += fabsf(Lr[x] - Lr[x + W]); gyr += fabsf(Rr[x] - Rr[x + W]); }
    }
    const float third = 1.f / 3.f;
    if (x < W - 1) {
      p2 = fabsf((dlv - dl[x + 1]) * expf(-gxl * third));
      p4 = fabsf((drv - dr[x + 1]) * expf(-gxr * third));
    }
    if (y < H - 1) {
      p3 = fabsf((dlv - dl[x + W]) * expf(-gyl * third));
      p5 = fabsf((drv - dr[x + W]) * expf(-gyr * third));
    }
    p6 = fabsf(sample1d(dr, W, pxl) - dlv);  // |gen_left(dr, dl) - dl|
    p7 = fabsf(sample1d(dl, W, pxr) - drv);  // |gen_right(dl, dr) - dr|
  }
  float vals[8] = {p0, p1, p2, p3, p4, p5, p6, p7};
  const int slot[8] = {0, 1, 4, 5, 6, 7, 8, 9};
  for (int j = 0; j < 8; ++j) {
    red[tid] = vals[j];
    for (int offr = 128; offr > 0; offr >>= 1) {
      __syncthreads();
      if (tid < offr) red[tid] += red[tid + offr];
    }
    __syncthreads();
    if (tid == 0) atomicAdd(&acc[slot[j]], red[0]);
    __syncthreads();
  }
}

// ---------------------------------------------------------------------------
// SSIM kernel. One wave32 per (b, ch, 14x16 output tile). The warped estimate
// is recomputed into LDS; all five pooled quantities {x, y, x^2, y^2, xy} are
// materialized as 16x20 planes (K padded 18 -> 20 with zeros so the WMMA
// A-operand loads are unguarded, aligned ds_load_b64). The horizontal 3-tap
// box sums are banded-matrix products with chained V_WMMA_F32_16X16X4_F32;
// the vertical 3-tap + SSIM formula run in VALU from LDS.
// acc slots: 2 ssim_l, 3 ssim_r.
// ---------------------------------------------------------------------------
__global__ void __launch_bounds__(32) ssim_kernel(
    const float* __restrict__ Lp, const float* __restrict__ Rp,
    const float* __restrict__ disp, float* __restrict__ acc,
    int H, int W, int tilesY, int tilesX) {
  __shared__ __align__(16) float P[5][16][20];   // quantity planes, K-padded
  __shared__ __align__(16) float Hh[5][16][16];  // horizontally pooled sums
  __shared__ float red[32];

  int bi = blockIdx.x;
  int tx = bi % tilesX; bi /= tilesX;
  int ty = bi % tilesY; bi /= tilesY;
  int ch = bi % 3;
  int b  = bi / 3;
  int oy = ty * 14, ox = tx * 16;
  int lane = threadIdx.x;
  int m    = lane & 15;   // M (A rows) / N (B cols) index for this lane
  int half = lane >> 4;

  size_t hw = (size_t)H * W;
  const float* Lc  = Lp + ((size_t)b * 3 + ch) * hw;
  const float* Rc  = Rp + ((size_t)b * 3 + ch) * hw;
  const float* dlm = disp + (size_t)b * 2 * hw;
  const float* drm = dlm + hw;

  // Band matrix T[k][n] = 1 iff k in {n, n+1, n+2}; B-operand slices are
  // invariant over quantity and pair -> build once, keep in VGPRs.
  // f32 4x16 B layout: VGPR0 holds K=4s (lanes 0-15) / K=4s+2 (lanes 16-31),
  // VGPR1 holds K=4s+1 / K=4s+3; i.e. this lane supplies K=ka, ka+1 at N=m.
  v2f bsl[5];
  #pragma unroll
  for (int s = 0; s < 5; ++s) {
    int ka = 4 * s + 2 * half;
    bsl[s].x = (ka     < 18 && ka     >= m && ka     <= m + 2) ? 1.f : 0.f;
    bsl[s].y = (ka + 1 < 18 && ka + 1 >= m && ka + 1 <= m + 2) ? 1.f : 0.f;
  }

  for (int pair = 0; pair < 2; ++pair) {
    const float* src = (pair == 0) ? Rc : Lc;   // warp source
    const float* yim = (pair == 0) ? Lc : Rc;   // reference image
    const float* dm  = (pair == 0) ? dlm : drm;
    float sgn = (pair == 0) ? -1.f : 1.f;

    // Fill 16x20 quantity planes (cols 18,19 are K-padding zeros).
    // 320 elements / 32 lanes = 10 each. Clamped loads only feed masked
    // outputs (valid output (r,c) needs inputs <= H-1 / W-1 only when valid).
    for (int k = 0; k < 10; ++k) {
      int e = lane + k * 32;
      int r = e / 20, c = e % 20;
      float xv = 0.f, yv = 0.f;
      if (c < 18) {
        int gy = oy + r; if (gy > H - 1) gy = H - 1;
        int gx = ox + c; if (gx > W - 1) gx = W - 1;
        float d  = dm[(size_t)gy * W + gx];
        float px = (float)gx + sgn * d * (float)(W - 1);
        xv = sample1d(src + (size_t)gy * W, W, px);
        yv = yim[(size_t)gy * W + gx];
      }
      P[0][r][c] = xv;
      P[1][r][c] = yv;
      P[2][r][c] = xv * xv;
      P[3][r][c] = yv * yv;
      P[4][r][c] = xv * yv;
    }
    __syncthreads();

    // Horizontal 3-tap box sum via WMMA: H = Q(16x20) * T(20x16).
    // f32 16x4 A layout: this lane supplies (M=m, K=ka) and (M=m, K=ka+1);
    // ka even and row stride 80B -> each slice is one aligned ds_load_b64.
    for (int q = 0; q < 5; ++q) {
      v2f asl[5];
      #pragma unroll
      for (int s = 0; s < 5; ++s)
        asl[s] = *reinterpret_cast<const v2f*>(&P[q][m][4 * s + 2 * half]);
      v8f cacc = {0.f, 0.f, 0.f, 0.f, 0.f, 0.f, 0.f, 0.f};
      #pragma unroll
      for (int s = 0; s < 5; ++s)
        cacc = wmma_f32_16x16x4(asl[s], bsl[s], cacc);
      // D layout: lanes 0-15 -> rows 0-7 (col m); lanes 16-31 -> rows 8-15.
      #pragma unroll
      for (int j = 0; j < 8; ++j) Hh[q][half * 8 + j][m] = cacc[j];
    }
    __syncthreads();

    // Vertical 3-tap + SSIM formula. 14x16 = 224 outputs, 7 per lane.
    float lsum = 0.f;
    for (int t = 0; t < 7; ++t) {
      int e = lane * 7 + t;
      int r = e >> 4, c = e & 15;
      if (oy + r <= H - 3 && ox + c <= W - 3) {
        float Sx  = Hh[0][r][c] + Hh[0][r + 1][c] + Hh[0][r + 2][c];
        float Sy  = Hh[1][r][c] + Hh[1][r + 1][c] + Hh[1][r + 2][c];
        float Sxx = Hh[2][r][c] + Hh[2][r + 1][c] + Hh[2][r + 2][c];
        float Syy = Hh[3][r][c] + Hh[3][r + 1][c] + Hh[3][r + 2][c];
        float Sxy = Hh[4][r][c] + Hh[4][r + 1][c] + Hh[4][r + 2][c];
        const float inv9 = 1.f / 9.f;
        float mux = Sx * inv9, muy = Sy * inv9;
        float sgx = Sxx * inv9 - mux * mux;
        float sgy = Syy * inv9 - muy * muy;
        float sxy = Sxy * inv9 - mux * muy;
        float nn = (2.f * mux * muy + 1e-4f) * (2.f * sxy + 9e-4f);
        float dd = (mux * mux + muy * muy + 1e-4f) * (sgx + sgy + 9e-4f);
        float v  = (1.f - nn / dd) * 0.5f;
        v = fminf(fmaxf(v, 0.f), 1.f);
        lsum += v;
      }
    }
    red[lane] = lsum;
    for (int offr = 16; offr > 0; offr >>= 1) {
      __syncthreads();
      if (lane < offr) red[lane] += red[lane + offr];
    }
    __syncthreads();
    if (lane == 0) atomicAdd(&acc[2 + pair], red[0]);
    __syncthreads();
  }
}

// ---------------------------------------------------------------------------
// Combine the 40 accumulator slots with the per-term mean normalizations and
// loss weights (ALPHA=0.85, W_GRAD=0.1, W_LR=1.0).
// ---------------------------------------------------------------------------
__global__ void finalize_kernel(const float* __restrict__ acc, float* __restrict__ out) {
  float total = 0.f;
  for (int i = 0; i < 4; ++i) {
    int H = 256 >> i, W = 512 >> i;
    const float* a = acc + 10 * i;
    float N1  = 16.f * 3.f * (float)H * (float)W;
    float Ns  = 16.f * 3.f * (float)(H - 2) * (float)(W - 2);
    float Nx  = 16.f * (float)H * (float)(W - 1);
    float Ny  = 16.f * (float)(H - 1) * (float)W;
    float Nlr = 16.f * (float)H * (float)W;
    total += 0.85f * (a[2] + a[3]) / Ns
           + 0.15f * (a[0] + a[1]) / N1
           + (0.1f / (float)(1 << i)) * ((a[4] + a[6]) / Nx + (a[5] + a[7]) / Ny)
           + (a[8] + a[9]) / Nlr;
  }
  out[0] = total;
}

// ---------------------------------------------------------------------------
extern "C" void kernel_launch(void* const* d_in, const int* in_sizes, int n_in,
                              void* d_out, int out_size, void* d_ws, size_t ws_size,
                              hipStream_t stream) {
  (void)in_sizes; (void)n_in; (void)out_size; (void)ws_size;
  const float* left  = (const float*)d_in[0];
  const float* right = (const float*)d_in[1];
  const float* disp[4] = {(const float*)d_in[2], (const float*)d_in[3],
                          (const float*)d_in[4], (const float*)d_in[5]};
  float* ws  = (float*)d_ws;
  float* acc = ws;  // 64 floats reserved; 40 used

  // Workspace pyramid buffers (levels 1..3 for left and right).
  size_t lvlN[4] = {0, (size_t)16 * 3 * 128 * 256, (size_t)16 * 3 * 64 * 128,
                    (size_t)16 * 3 * 32 * 64};
  size_t off = 64;
  float* Lw[4] = {nullptr, nullptr, nullptr, nullptr};
  float* Rw[4] = {nullptr, nullptr, nullptr, nullptr};
  for (int l = 1; l < 4; ++l) { Lw[l] = ws + off; off += lvlN[l]; }
  for (int l = 1; l < 4; ++l) { Rw[l] = ws + off; off += lvlN[l]; }
  const float* Lp[4] = {left, Lw[1], Lw[2], Lw[3]};
  const float* Rp[4] = {right, Rw[1], Rw[2], Rw[3]};

  init_kernel<<<1, 64, 0, stream>>>(acc);

  for (int l = 1; l < 4; ++l) {
    int h = 256 >> l, w = 512 >> l, s = 1 << l;
    int n = 16 * 3 * h * w;
    int nb = (n + 255) / 256;
    pyr_kernel<<<nb, 256, 0, stream>>>(left,  Lw[l], s, h, w, n);
    pyr_kernel<<<nb, 256, 0, stream>>>(right, Rw[l], s, h, w, n);
  }

  for (int i = 0; i < 4; ++i) {
    int H = 256 >> i, W = 512 >> i;
    int n = 16 * H * W;
    warp_loss_kernel<<<(n + 255) / 256, 256, 0, stream>>>(
        Lp[i], Rp[i], disp[i], acc + 10 * i, H, W, n);
    int ty = (H - 2 + 13) / 14;
    int tx = (W - 2 + 15) / 16;
    ssim_kernel<<<16 * 3 * ty * tx, 32, 0, stream>>>(
        Lp[i], Rp[i], disp[i], acc + 10 * i, H, W, ty, tx);
  }

  finalize_kernel<<<1, 1, 0, stream>>>(acc, (float*)d_out);
}